// EncoderBlock_45500883534462
// MI455X (gfx1250) — compile-verified
//
#include <hip/hip_runtime.h>
#include <math.h>

typedef __attribute__((ext_vector_type(2))) float v2f;
typedef __attribute__((ext_vector_type(8))) float v8f;

#define N_B     8
#define IN_C    64
#define Hh      256
#define Ww      256
#define HW      65536
#define OUT_C   128
#define NUM_SEL 48
#define PW_OUT  112
#define BN_EPS  1e-5f
#define OUT_ELEMS  (N_B * OUT_C * 128 * 128)   /* 16777216 */
#define SKIP_ELEMS (N_B * IN_C * Hh * Ww)      /* 33554432 */
#define KSPLIT  64
#define KCH     64
#define LSTR    68

#if __has_builtin(__builtin_amdgcn_tensor_load_to_lds)
#define HAVE_TDM 1
typedef unsigned int u32x4 __attribute__((ext_vector_type(4)));
typedef int          i32x4 __attribute__((ext_vector_type(4)));
typedef int          i32x8 __attribute__((ext_vector_type(8)));

// TDM: DMA one 64-row x 64-float f32 chunk (row stride 65536 floats in global)
// into LDS, hardware-padded +4 DWORDs per 64-DWORD row -> 68-float LDS stride.
__device__ __forceinline__ void tdm_load_chunk(const float* gsrc, unsigned lds_byte_addr)
{
    unsigned long long ga = (unsigned long long)(size_t)gsrc;
    u32x4 g0;
    g0[0] = 1u;                                         // count=1 (valid descriptor)
    g0[1] = lds_byte_addr;                              // lds_addr
    g0[2] = (unsigned)(ga & 0xffffffffu);               // global_addr[31:0]
    g0[3] = (unsigned)((ga >> 32) & 0x01ffffffu)        // global_addr[56:32]
          | (2u << 30);                                 // type=2 ("image")
    i32x8 g1;
    g1[0] = (2 << 16)        // data_size = 4B
          | (1 << 20)        // pad_enable
          | (5 << 22)        // pad_interval: 64 DWORDs
          | (3 << 25);       // pad_amount:   4 DWORDs
    g1[1] = 0;                                          // atomic_addr=0, tdim0[15:0]=0
    g1[2] = (64 << 16) | 1;                             // tdim0[31:16]=1 (65536), tdim1 lo=64
    g1[3] = (64 << 16);                                 // tdim1 hi=0, tile_dim0=64
    g1[4] = 64;                                         // tile_dim1=64, tile_dim2=0
    g1[5] = 65536;                                      // tensor_dim0_stride[31:0]
    g1[6] = 0;                                          // stride0 hi, stride1 lo
    g1[7] = 0;
    i32x4 gz = {0, 0, 0, 0};
#if __has_include(<hip/amd_detail/amd_gfx1250_TDM.h>)
    i32x8 gz8 = {0, 0, 0, 0, 0, 0, 0, 0};               // clang-23 toolchain: 6-arg form
    __builtin_amdgcn_tensor_load_to_lds(g0, g1, gz, gz, gz8, 0);
#else
    __builtin_amdgcn_tensor_load_to_lds(g0, g1, gz, gz, 0);   // ROCm 7.2: 5-arg form
#endif
}
#endif

// ---------------------------------------------------------------------------
// k1: skip = BN(x + dw3x1(x) + dw1x3(x)); 4 pixels per thread, float4 I/O.
// ---------------------------------------------------------------------------
__global__ void k1_skip(const float* __restrict__ x,
                        const float* __restrict__ dwh_w, const float* __restrict__ dwh_b,
                        const float* __restrict__ dww_w, const float* __restrict__ dww_b,
                        const float* __restrict__ bn_g, const float* __restrict__ bn_b,
                        const float* __restrict__ bn_m, const float* __restrict__ bn_v,
                        float* __restrict__ skip)
{
    int t  = blockIdx.x * blockDim.x + threadIdx.x;     // 8,388,608 threads
    int x4 = t & 63;
    int y  = (t >> 6) & 255;
    int c  = (t >> 14) & 63;
    int n  = t >> 20;

    const float* row = x + (((size_t)(n * IN_C + c)) << 16) + (size_t)y * Ww;
    float4 xc = ((const float4*)row)[x4];
    float4 up = make_float4(0.f, 0.f, 0.f, 0.f);
    float4 dn = up;
    if (y > 0)   up = *(const float4*)(row - Ww + x4 * 4);
    if (y < 255) dn = *(const float4*)(row + Ww + x4 * 4);
    float lft = (x4 > 0)  ? row[x4 * 4 - 1] : 0.f;
    float rgt = (x4 < 63) ? row[x4 * 4 + 4] : 0.f;

    float wh0 = dwh_w[c*3+0], wh1 = dwh_w[c*3+1], wh2 = dwh_w[c*3+2], bh = dwh_b[c];
    float w0  = dww_w[c*3+0], w1  = dww_w[c*3+1], w2  = dww_w[c*3+2], bw = dww_b[c];
    float inv  = bn_g[c] * rsqrtf(bn_v[c] + BN_EPS);
    float beta = bn_b[c] - bn_m[c] * inv;

    float cc[4] = {xc.x, xc.y, xc.z, xc.w};
    float uu[4] = {up.x, up.y, up.z, up.w};
    float dd[4] = {dn.x, dn.y, dn.z, dn.w};
    float ln[4] = {lft,  xc.x, xc.y, xc.z};
    float rn[4] = {xc.y, xc.z, xc.w, rgt};

    float4 o;
    float* op = (float*)&o;
#pragma unroll
    for (int i = 0; i < 4; ++i) {
        float h = cc[i]
                + (wh0 * uu[i] + wh1 * cc[i] + wh2 * dd[i] + bh)
                + (w0  * ln[i] + w1  * cc[i] + w2  * rn[i] + bw);
        op[i] = h * inv + beta;
    }
    *(float4*)(skip + (((size_t)(n * IN_C + c)) << 16) + (size_t)y * Ww + x4 * 4) = o;
}

// ---------------------------------------------------------------------------
// k2: Gram G[n] = S S^T (S = skip[n] as 64 x 65536) with V_WMMA_F32_16X16X4.
// grid (KSPLIT, N_B), 512 threads = 16 waves, one 16x16 tile each (ti<=tj).
// TDM path: double-buffered LDS; wave 0 issues tensor_load_to_lds for chunk
// c+1 while all waves run 16 WMMAs on chunk c; s_wait_tensorcnt + barrier.
// Fallback path: synchronous float4 staging (stride 68, conflict-free).
// ---------------------------------------------------------------------------
__global__ void k2_gram(const float* __restrict__ skip, float* __restrict__ G)
{
#if HAVE_TDM
    __shared__ float lds[2][IN_C * LSTR];
#else
    __shared__ float lds[1][IN_C * LSTR];
#endif
    int n    = blockIdx.y;
    int kb0  = blockIdx.x * (HW / KSPLIT);          // 1024-wide K slice
    int tid  = threadIdx.x;
    int lane = tid & 31;
    int wave = tid >> 5;
    int ti = wave >> 2, tj = wave & 3;
    bool active = (ti <= tj);                        // symmetric: upper+diag only

    int mrow = ti * 16 + (lane & 15);
    int nrow = tj * 16 + (lane & 15);
    int koff = (lane >> 4) << 1;                     // A/B per-lane K offset

    const float* sb = skip + (((size_t)n * IN_C) << 16);
    v8f acc = {0.f, 0.f, 0.f, 0.f, 0.f, 0.f, 0.f, 0.f};

#if HAVE_TDM
    unsigned ldsBase  = (unsigned)(size_t)&lds[0][0];
    const unsigned bufBytes = IN_C * LSTR * 4;
    if (wave == 0) {
        tdm_load_chunk(sb + kb0, ldsBase);           // prime chunk 0
        __builtin_amdgcn_s_wait_tensorcnt(0);
    }
    __syncthreads();
    for (int ch = 0; ch < (HW / KSPLIT) / KCH; ++ch) {       // 16 chunks
        if (wave == 0 && ch + 1 < (HW / KSPLIT) / KCH)
            tdm_load_chunk(sb + kb0 + (ch + 1) * KCH,        // prefetch chunk c+1
                           ldsBase + ((unsigned)((ch + 1) & 1)) * bufBytes);
        const float* bufp = &lds[ch & 1][0];
        if (active) {
#pragma unroll
            for (int kk = 0; kk < KCH; kk += 4) {
                v2f a = *(const v2f*)&bufp[mrow * LSTR + kk + koff];
                v2f b = *(const v2f*)&bufp[nrow * LSTR + kk + koff];
                acc = __builtin_amdgcn_wmma_f32_16x16x4_f32(
                          false, a, false, b, (short)0, acc, false, false);
            }
        }
        if (wave == 0)
            __builtin_amdgcn_s_wait_tensorcnt(0);    // chunk c+1 landed in LDS
        __syncthreads();
    }
#else
    int lrow  = tid >> 3;                            // cooperative loader: row
    int lkoff = (tid & 7) * 8;                       // 8 floats per thread
    for (int ch = 0; ch < (HW / KSPLIT) / KCH; ++ch) {       // 16 chunks
        int kb = kb0 + ch * KCH;
        const float4* src = (const float4*)(sb + ((size_t)lrow << 16) + kb + lkoff);
        __builtin_prefetch(src + 16, 0, 1);          // next chunk -> global_prefetch
        float4 a0 = src[0], a1 = src[1];
        *(float4*)&lds[0][lrow * LSTR + lkoff]     = a0;
        *(float4*)&lds[0][lrow * LSTR + lkoff + 4] = a1;
        __syncthreads();
        if (active) {
#pragma unroll
            for (int kk = 0; kk < KCH; kk += 4) {
                v2f a = *(const v2f*)&lds[0][mrow * LSTR + kk + koff];
                v2f b = *(const v2f*)&lds[0][nrow * LSTR + kk + koff];
                acc = __builtin_amdgcn_wmma_f32_16x16x4_f32(
                          false, a, false, b, (short)0, acc, false, false);
            }
        }
        __syncthreads();
    }
#endif

    if (active) {
        int Mb = ti * 16 + (lane >> 4) * 8;          // C/D layout: M = r + 8*(lane>>4)
        int Nc = tj * 16 + (lane & 15);              //             N = lane & 15
#pragma unroll
        for (int r = 0; r < 8; ++r)
            atomicAdd(&G[((n << 6) + Mb + r) * 64 + Nc], acc[r]);
    }
}

// ---------------------------------------------------------------------------
// k3: per-batch: norms from diag(G), write `upper`, count, top-k(48) with
// lax.top_k tie-breaking (count desc, index asc), emit sel/unsel index lists.
// ---------------------------------------------------------------------------
__global__ void k3_select(const float* __restrict__ G, float* __restrict__ upper,
                          int* __restrict__ selIdx, int* __restrict__ unselIdx)
{
    __shared__ float snrm[64];
    __shared__ int scount[64];
    __shared__ int smask[64];
    int n = blockIdx.x;
    int j = threadIdx.x;
    const float* Gn = G + (n << 12);

    snrm[j] = fmaxf(sqrtf(fmaxf(Gn[j * 64 + j], 0.f)), 1e-12f);
    __syncthreads();

    int cnt = 64 - j;                                 // zeros of triu (i>=j) are < 0.5
    float nj = snrm[j];
    for (int i = 0; i < 64; ++i) {
        float u = 0.f;
        if (i < j) {
            u = Gn[i * 64 + j] / (snrm[i] * nj);
            if (u < 0.5f) cnt++;
        }
        upper[(n << 12) + i * 64 + j] = u;
    }
    scount[j] = cnt;
    __syncthreads();

    int rank = 0;
    for (int i = 0; i < 64; ++i) {
        int ci = scount[i];
        if (ci > cnt || (ci == cnt && i < j)) rank++;
    }
    smask[j] = (rank < NUM_SEL) ? 1 : 0;
    __syncthreads();

    if (j == 0) {                                     // stable argsort of mask
        int si = 0, ui = 0;
        for (int c = 0; c < 64; ++c) {
            if (smask[c]) selIdx[n * NUM_SEL + si++] = c;
            else          unselIdx[n * 16 + ui++]    = c;
        }
    }
}

// ---------------------------------------------------------------------------
// k4: pw conv (112x48 @ gathered 48xHW) via WMMA f32 16x16x4, fused with
// +bias, BN1, 2x2 maxpool (vert in-register, horiz via shfl_xor), exact GELU.
// ---------------------------------------------------------------------------
__global__ void k4_pw(const float* __restrict__ skip, const float* __restrict__ pw_w,
                      const float* __restrict__ pw_b,
                      const float* __restrict__ g1, const float* __restrict__ b1,
                      const float* __restrict__ m1, const float* __restrict__ v1,
                      const int* __restrict__ selIdx, float* __restrict__ out)
{
    __shared__ float sW[16 * 48];
    __shared__ int   sIdx[48];
    __shared__ float sInv[16], sBeta[16], sPwb[16];

    int mt  = blockIdx.y;
    int n   = blockIdx.z;
    int tid = threadIdx.x;

    for (int i = tid; i < 16 * 48; i += 256) sW[i] = pw_w[mt * 768 + i];
    if (tid < 48) sIdx[tid] = selIdx[n * NUM_SEL + tid];
    if (tid < 16) {
        int ch = mt * 16 + tid;
        float inv = g1[ch] * rsqrtf(v1[ch] + BN_EPS);
        sInv[tid]  = inv;
        sBeta[tid] = b1[ch] - m1[ch] * inv;
        sPwb[tid]  = pw_b[ch];
    }
    __syncthreads();

    int lane = tid & 31;
    int wave = tid >> 5;
    int task = blockIdx.x * 8 + wave;                 // 2048 tasks per (n,mt)
    int yo   = task >> 4;                             // pooled row 0..127
    int xb   = (task & 15) << 4;                      // input x base

    const float* sk = skip + (((size_t)n * IN_C) << 16);
    int y0   = yo << 1;
    int posT = y0 * Ww + xb + (lane & 15);
    int posB = posT + Ww;
    int koff = (lane >> 4) << 1;
    int arow = (lane & 15) * 48;

    v8f accT = {0.f, 0.f, 0.f, 0.f, 0.f, 0.f, 0.f, 0.f};
    v8f accB = accT;
#pragma unroll
    for (int kb = 0; kb < 48; kb += 4) {
        int c0 = sIdx[kb + koff];
        int c1 = sIdx[kb + koff + 1];
        v2f a = *(const v2f*)&sW[arow + kb + koff];
        const float* p0 = sk + ((size_t)c0 << 16);
        const float* p1 = sk + ((size_t)c1 << 16);
        v2f bt, bb;
        bt[0] = p0[posT]; bt[1] = p1[posT];
        bb[0] = p0[posB]; bb[1] = p1[posB];
        accT = __builtin_amdgcn_wmma_f32_16x16x4_f32(
                   false, a, false, bt, (short)0, accT, false, false);
        accB = __builtin_amdgcn_wmma_f32_16x16x4_f32(
                   false, a, false, bb, (short)0, accB, false, false);
    }

    int half   = lane >> 4;
    int xo     = (xb >> 1) + ((lane & 15) >> 1);
    bool writer = ((lane & 1) == 0);
#pragma unroll
    for (int r = 0; r < 8; ++r) {
        float m      = fmaxf(accT[r], accB[r]);       // vertical pool
        float other  = __shfl_xor(m, 1, 32);          // horizontal pool (wave32)
        float pooled = fmaxf(m, other);
        int   cr = r + 8 * half;
        float z  = (pooled + sPwb[cr]) * sInv[cr] + sBeta[cr];
        float ge = 0.5f * z * (1.0f + erff(z * 0.70710678118654752f));
        if (writer) {
            int ch = mt * 16 + cr;
            out[(((size_t)(n * OUT_C + ch)) << 14) + yo * 128 + xo] = ge;
        }
    }
}

// ---------------------------------------------------------------------------
// k5: unselected 16 channels -> cat channels 112..127: pool + BN1 + GELU.
// ---------------------------------------------------------------------------
__global__ void k5_unsel(const float* __restrict__ skip,
                         const float* __restrict__ g1, const float* __restrict__ b1,
                         const float* __restrict__ m1, const float* __restrict__ v1,
                         const int* __restrict__ unselIdx, float* __restrict__ out)
{
    int t  = blockIdx.x * blockDim.x + threadIdx.x;   // 2,097,152 threads
    int xo = t & 127;
    int yo = (t >> 7) & 127;
    int u  = (t >> 14) & 15;
    int n  = t >> 18;

    int c = unselIdx[n * 16 + u];
    const float* p = skip + (((size_t)(n * IN_C + c)) << 16)
                   + (size_t)(yo * 2) * Ww + xo * 2;
    float m = fmaxf(fmaxf(p[0], p[1]), fmaxf(p[Ww], p[Ww + 1]));

    int ch = PW_OUT + u;
    float inv = g1[ch] * rsqrtf(v1[ch] + BN_EPS);
    float z   = m * inv + (b1[ch] - m1[ch] * inv);
    float ge  = 0.5f * z * (1.0f + erff(z * 0.70710678118654752f));
    out[(((size_t)(n * OUT_C + ch)) << 14) + yo * 128 + xo] = ge;
}

// ---------------------------------------------------------------------------
extern "C" void kernel_launch(void* const* d_in, const int* in_sizes, int n_in,
                              void* d_out, int out_size, void* d_ws, size_t ws_size,
                              hipStream_t stream)
{
    const float* x     = (const float*)d_in[0];
    const float* dwh_w = (const float*)d_in[1];
    const float* dwh_b = (const float*)d_in[2];
    const float* dww_w = (const float*)d_in[3];
    const float* dww_b = (const float*)d_in[4];
    const float* bn_g  = (const float*)d_in[5];
    const float* bn_b  = (const float*)d_in[6];
    const float* bn_m  = (const float*)d_in[7];
    const float* bn_v  = (const float*)d_in[8];
    const float* pw_w  = (const float*)d_in[9];
    const float* pw_b  = (const float*)d_in[10];
    const float* bn1_g = (const float*)d_in[11];
    const float* bn1_b = (const float*)d_in[12];
    const float* bn1_m = (const float*)d_in[13];
    const float* bn1_v = (const float*)d_in[14];

    float* outp  = (float*)d_out;              // (8,128,128,128)
    float* skip  = outp + OUT_ELEMS;           // (8,64,256,256)
    float* upper = skip + SKIP_ELEMS;          // (8,64,64)

    float* G         = (float*)d_ws;                       // 8*64*64 f32
    int*   selIdx    = (int*)(G + N_B * 64 * 64);          // 8*48 i32
    int*   unselIdx  = selIdx + N_B * NUM_SEL;             // 8*16 i32

    hipMemsetAsync(G, 0, N_B * 64 * 64 * sizeof(float), stream);

    k1_skip<<<32768, 256, 0, stream>>>(x, dwh_w, dwh_b, dww_w, dww_b,
                                       bn_g, bn_b, bn_m, bn_v, skip);
    k2_gram<<<dim3(KSPLIT, N_B), 512, 0, stream>>>(skip, G);
    k3_select<<<N_B, 64, 0, stream>>>(G, upper, selIdx, unselIdx);
    k4_pw<<<dim3(256, 7, N_B), 256, 0, stream>>>(skip, pw_w, pw_b,
                                                 bn1_g, bn1_b, bn1_m, bn1_v,
                                                 selIdx, outp);
    k5_unsel<<<8192, 256, 0, stream>>>(skip, bn1_g, bn1_b, bn1_m, bn1_v,
                                       unselIdx, outp);
}